// DCN_Conv_31035433680960
// MI455X (gfx1250) — compile-verified
//
#include <hip/hip_runtime.h>
#include <cmath>

// ---------------------------------------------------------------------------
// Dynamic Snake Conv 3D for MI455X (gfx1250, wave32, WMMA).
// fp32 end-to-end: ~100MB traffic / ~6.3 GFLOP -> memory bound at 23.3 TB/s,
// so full precision via V_WMMA_F32_16X16X4_F32 for both implicit GEMMs.
// B matrices are pre-packed into WMMA fragment order so the GEMM inner loop
// is: ds_load_2addr(A) + coalesced global_load_b64(B) + v_wmma.
// ---------------------------------------------------------------------------

typedef __attribute__((ext_vector_type(2))) float v2f;
typedef __attribute__((ext_vector_type(4))) float v4f;
typedef __attribute__((ext_vector_type(8))) float v8f;
typedef __attribute__((ext_vector_type(4))) int   v4i;

// Address-space-qualified pointer types for the async LDS-DMA builtin
// (param types revealed by the round-2 diagnostic: v4i* in AS1 / AS3).
typedef __attribute__((address_space(1))) v4i* gas_b128_t;
typedef __attribute__((address_space(3))) v4i* las_b128_t;

#define DD    48
#define CIN   16
#define COUT  32
#define KPTS  9
#define VOL   (DD * DD * DD)          // 110592
#define OFFCH 54                      // 6*K offset channels
#define EPSN  1e-5f
#define DEFSTR 148                    // sdef row stride (bank-conflict pad)

#define NB1   (4 * 27 * 4 * 32)       // wB1 fragments: [coTile][tap][cb4][lane]
#define NB2   (2 * 36 * 32)           // wB2 fragments: [nt][kb4][lane]

// ---------------------------------------------------------------------------
// Kernel 0: pack both conv weight tensors into WMMA B-fragment order (v2f per
// lane, zero-padded for co >= 54).  Runs once; panels live hot in L2/L0.
// ---------------------------------------------------------------------------
__global__ __launch_bounds__(256) void k_pack_w(
    const float* __restrict__ ow, const float* __restrict__ dwp,
    v2f* __restrict__ wB1, v2f* __restrict__ wB2)
{
    const int t = blockIdx.x * 256 + threadIdx.x;
    const int stride = gridDim.x * 256;

    for (int i = t; i < NB1; i += stride) {
        const int lane = i & 31;
        int r = i >> 5;
        const int cb4 = r & 3;  r >>= 2;
        const int tap = r % 27;
        const int wt  = r / 27;
        const int co  = wt * 16 + (lane & 15);
        const int ci  = cb4 * 4 + ((lane >> 4) << 1);
        v2f v;
        v.x = (co < OFFCH) ? ow[(co * CIN + ci) * 27 + tap]     : 0.f;
        v.y = (co < OFFCH) ? ow[(co * CIN + ci + 1) * 27 + tap] : 0.f;
        wB1[i] = v;
    }
    for (int i = t; i < NB2; i += stride) {
        const int lane = i & 31;
        int r = i >> 5;
        const int kb4 = r % 36;
        const int nt  = r / 36;
        const int co  = nt * 16 + (lane & 15);
        const int k0  = kb4 * 4 + ((lane >> 4) << 1);
        v2f v;
        v.x = dwp[(co * CIN + (k0 & 15)) * KPTS + (k0 >> 4)];
        v.y = dwp[(co * CIN + ((k0 + 1) & 15)) * KPTS + ((k0 + 1) >> 4)];
        wB2[i] = v;
    }
}

// ---------------------------------------------------------------------------
// Kernel 1: offset conv 3x3x3 (16 -> 54ch) + BN(eval) + tanh, implicit GEMM.
// Block = 128 threads = 4 waves; wave w owns co-tile [16w,16w+16).
// 27 taps x 4 k-steps = 108 f32 WMMAs per wave.
// ---------------------------------------------------------------------------
__global__ __launch_bounds__(128) void k_offset_conv(
    const float* __restrict__ x,  const v2f* __restrict__ wB1,
    const float* __restrict__ ob, const float* __restrict__ bng,
    const float* __restrict__ bnb, const float* __restrict__ bnm,
    const float* __restrict__ bnv, float* __restrict__ off)
{
    __shared__ float sx[CIN * 3 * 3 * 18];   // [ci][dz][dw][hl] halo tile

    const int tid  = threadIdx.x;
    const int lane = tid & 31;
    const int wid  = tid >> 5;
    const int h0   = blockIdx.x * 16;
    const int w0   = blockIdx.y;
    const int d0   = blockIdx.z;

    for (int i = tid; i < CIN * 3 * 3 * 18; i += 128) {
        int hl = i % 18; int t = i / 18;
        int dwv = t % 3; t /= 3;
        int dzv = t % 3; int ci = t / 3;
        int dd = d0 + dzv - 1, ww = w0 + dwv - 1, hh = h0 + hl - 1;
        float val = 0.f;
        if ((unsigned)dd < DD && (unsigned)ww < DD && (unsigned)hh < DD)
            val = x[((ci * DD + dd) * DD + ww) * DD + hh];
        sx[i] = val;
    }
    __syncthreads();

    const int n    = lane & 15;
    const int half = lane >> 4;
    const int co   = wid * 16 + n;

    // B panel walks sequentially: tap-major, cb4-minor, 32 fragments apart
    const v2f* wb = wB1 + (size_t)(wid * 27 * 4) * 32 + lane;

    v8f acc = {};
    for (int dz = 0; dz < 3; ++dz)
    for (int dwv = 0; dwv < 3; ++dwv)
    for (int dh = 0; dh < 3; ++dh) {
        #pragma unroll
        for (int cb4 = 0; cb4 < 4; ++cb4) {
            const int ci0 = cb4 * 4 + half * 2;
            v2f a;
            a.x = sx[((ci0 * 3 + dz) * 3 + dwv) * 18 + n + dh];
            a.y = sx[(((ci0 + 1) * 3 + dz) * 3 + dwv) * 18 + n + dh];
            const v2f b = *wb;  wb += 32;
            acc = __builtin_amdgcn_wmma_f32_16x16x4_f32(
                      false, a, false, b, (short)0, acc, false, false);
        }
    }

    if (co < OFFCH) {
        const float sc = bng[co] * rsqrtf(bnv[co] + EPSN);
        const float sh = bnb[co] - bnm[co] * sc;
        const float bias = ob[co];
        #pragma unroll
        for (int r = 0; r < 8; ++r) {
            const int h = h0 + r + half * 8;
            float v = acc[r] + bias;
            v = tanhf(v * sc + sh);
            off[co * VOL + (d0 * DD + w0) * DD + h] = v;
        }
    }
}

// ---------------------------------------------------------------------------
// Kernel 2: snake coords + trilinear gather into LDS A-panel, fused
// (1,1,9)-stride snake conv as WMMA GEMM (M=48,N=32,K=144), plus block-level
// GroupNorm sum/sumsq (ds_add_f32 -> global_atomic_add_f32).
// Offset-channel row staged via GLOBAL_LOAD_ASYNC_TO_LDS_B128 (ASYNCcnt).
// ---------------------------------------------------------------------------
__global__ __launch_bounds__(192) void k_snake(
    const float* __restrict__ x,   const float* __restrict__ off,
    const v2f* __restrict__ wB2,   const float* __restrict__ db,
    float* __restrict__ pre, float* __restrict__ stats)
{
    __shared__ float sdef[DD][DEFSTR];   // A panel: [m=h][k = kk*16 + ci]
    __shared__ float soff[18 * DD];      // off[ch 0..17] for this (d,w) row
    __shared__ float gsum[8], gsq[8];

    const int tid  = threadIdx.x;
    const int lane = tid & 31;
    const int wid  = tid >> 5;
    const int w0   = blockIdx.x;
    const int d0   = blockIdx.y;
    const int rowbase = (d0 * DD + w0) * DD;

    if (tid < 8) { gsum[tid] = 0.f; gsq[tid] = 0.f; }
    if (tid == 0) __builtin_prefetch(wB2, 0, 0);     // warm B panel

    // ---- Async stage: 18 channels x 48 floats of `off` -> LDS (b128) ----
    for (int i = tid; i < 18 * (DD / 4); i += 192) {
        const int ch = i / (DD / 4), seg = i % (DD / 4);
        const float* src = off + (size_t)ch * VOL + rowbase + seg * 4;
        float* dst = &soff[ch * DD + seg * 4];
        __builtin_amdgcn_global_load_async_to_lds_b128(
            (gas_b128_t)(void*)src, (las_b128_t)(void*)dst, 0, 0);
    }
    asm volatile("s_wait_asynccnt 0x0" ::: "memory");
    __syncthreads();

    // ---- Phase 1: 48 voxels x 9 snake points -> sampled 16-ch vectors ----
    for (int idx = tid; idx < DD * KPTS; idx += 192) {
        const int m  = idx / KPTS;
        const int kk = idx % KPTS;

        float cz = 0.f, cy = 0.f;
        if (kk < 4) {
            for (int j = kk; j <= 3; ++j) {
                cz += soff[j * DD + m];
                cy += soff[(KPTS + j) * DD + m];
            }
        } else if (kk > 4) {
            for (int j = 5; j <= kk; ++j) {
                cz += soff[j * DD + m];
                cy += soff[(KPTS + j) * DD + m];
            }
        }
        float z  = fminf(fmaxf((float)d0 + cz, 0.f), (float)(DD - 1));
        float y  = fminf(fmaxf((float)w0 + cy, 0.f), (float)(DD - 1));
        float xx = fminf(fmaxf((float)m + (float)(kk - 4), 0.f), (float)(DD - 1));

        int z0 = min(max((int)floorf(z), 0), DD - 1), z1 = min(z0 + 1, DD - 1);
        int y0 = min(max((int)floorf(y), 0), DD - 1), y1 = min(y0 + 1, DD - 1);
        int x0 = min(max((int)floorf(xx), 0), DD - 1), x1 = min(x0 + 1, DD - 1);
        float wz0 = (float)z1 - z,  wz1 = z  - (float)z0;
        float wy0 = (float)y1 - y,  wy1 = y  - (float)y0;
        float wx0 = (float)x1 - xx, wx1 = xx - (float)x0;

        int ofs[8];
        float wgt[8];
        ofs[0] = (z0 * DD + y0) * DD + x0;  wgt[0] = wz0 * wy0 * wx0;
        ofs[1] = (z0 * DD + y0) * DD + x1;  wgt[1] = wz0 * wy0 * wx1;
        ofs[2] = (z0 * DD + y1) * DD + x0;  wgt[2] = wz0 * wy1 * wx0;
        ofs[3] = (z0 * DD + y1) * DD + x1;  wgt[3] = wz0 * wy1 * wx1;
        ofs[4] = (z1 * DD + y0) * DD + x0;  wgt[4] = wz1 * wy0 * wx0;
        ofs[5] = (z1 * DD + y0) * DD + x1;  wgt[5] = wz1 * wy0 * wx1;
        ofs[6] = (z1 * DD + y1) * DD + x0;  wgt[6] = wz1 * wy1 * wx0;
        ofs[7] = (z1 * DD + y1) * DD + x1;  wgt[7] = wz1 * wy1 * wx1;

        #pragma unroll 4
        for (int ci = 0; ci < CIN; ++ci) {
            const float* xc = x + ci * VOL;
            float s = 0.f;
            #pragma unroll
            for (int t = 0; t < 8; ++t) s += wgt[t] * xc[ofs[t]];
            sdef[m][kk * 16 + ci] = s;
        }
    }
    __syncthreads();

    // ---- Phase 2: WMMA GEMM, 3 M-subtiles x 2 N-subtiles on 6 waves ----
    if (wid < 6) {
        const int mt   = wid >> 1;
        const int nt   = wid & 1;
        const int n    = lane & 15;
        const int half = lane >> 4;
        const int co   = nt * 16 + n;

        const v2f* wb = wB2 + (size_t)(nt * 36) * 32 + lane;

        v8f acc = {};
        for (int kb4 = 0; kb4 < 36; ++kb4) {
            const int k0 = kb4 * 4 + half * 2;
            v2f a;
            a.x = sdef[mt * 16 + n][k0];
            a.y = sdef[mt * 16 + n][k0 + 1];
            const v2f b = *wb;  wb += 32;
            acc = __builtin_amdgcn_wmma_f32_16x16x4_f32(
                      false, a, false, b, (short)0, acc, false, false);
        }

        const float bias = db[co];
        float ls = 0.f, lq = 0.f;
        #pragma unroll
        for (int r = 0; r < 8; ++r) {
            const int h = mt * 16 + r + half * 8;
            const float v = acc[r] + bias;
            pre[co * VOL + rowbase + h] = v;
            ls += v; lq += v * v;
        }
        atomicAdd(&gsum[co >> 2], ls);   // ds_add_f32
        atomicAdd(&gsq[co >> 2], lq);
    }
    __syncthreads();

    if (tid < 8) {
        atomicAdd(&stats[tid], gsum[tid]);       // global_atomic_add_f32
        atomicAdd(&stats[8 + tid], gsq[tid]);
    }
}

// ---------------------------------------------------------------------------
// Kernel 3: GroupNorm finalize + ReLU, float4-vectorized bandwidth pass.
// ---------------------------------------------------------------------------
__global__ __launch_bounds__(256) void k_gn_relu(
    const v4f* __restrict__ pre, const float* __restrict__ stats,
    const float* __restrict__ gg,  const float* __restrict__ gb,
    v4f* __restrict__ out)
{
    const int i = blockIdx.x * 256 + threadIdx.x;        // float4 index
    if (i >= COUT * (VOL / 4)) return;
    const int co = i / (VOL / 4);                        // VOL % 4 == 0
    const int g  = co >> 2;
    const float cnt = 4.0f * (float)VOL;
    const float mu  = stats[g] / cnt;
    const float var = stats[8 + g] / cnt - mu * mu;
    const float rs  = rsqrtf(var + EPSN) * gg[co];
    const float sh  = gb[co] - mu * rs;
    v4f v = pre[i];
    v4f o;
    o.x = fmaxf(v.x * rs + sh, 0.f);
    o.y = fmaxf(v.y * rs + sh, 0.f);
    o.z = fmaxf(v.z * rs + sh, 0.f);
    o.w = fmaxf(v.w * rs + sh, 0.f);
    out[i] = o;
}

// ---------------------------------------------------------------------------
extern "C" void kernel_launch(void* const* d_in, const int* in_sizes, int n_in,
                              void* d_out, int out_size, void* d_ws, size_t ws_size,
                              hipStream_t stream)
{
    const float* x   = (const float*)d_in[0];
    const float* ow  = (const float*)d_in[1];
    const float* ob  = (const float*)d_in[2];
    const float* bng = (const float*)d_in[3];
    const float* bnb = (const float*)d_in[4];
    const float* bnm = (const float*)d_in[5];
    const float* bnv = (const float*)d_in[6];
    const float* dw  = (const float*)d_in[7];
    const float* db  = (const float*)d_in[8];
    const float* gng = (const float*)d_in[9];
    const float* gnb = (const float*)d_in[10];

    float* off   = (float*)d_ws;               // 54 * VOL
    float* pre   = off + (size_t)OFFCH * VOL;  // 32 * VOL
    float* stats = pre + (size_t)COUT * VOL;   // 16 floats
    v2f*   wB1   = (v2f*)(stats + 16);         // NB1 v2f
    v2f*   wB2   = wB1 + NB1;                  // NB2 v2f

    (void)hipMemsetAsync(stats, 0, 16 * sizeof(float), stream);

    k_pack_w<<<16, 256, 0, stream>>>(ow, dw, wB1, wB2);

    dim3 g1(DD / 16, DD, DD);                  // (h-tiles, w, d)
    k_offset_conv<<<g1, 128, 0, stream>>>(x, wB1, ob, bng, bnb, bnm, bnv, off);

    dim3 g2(DD, DD);                           // (w, d)
    k_snake<<<g2, 192, 0, stream>>>(x, off, wB2, db, pre, stats);

    const int n4 = COUT * (VOL / 4);
    k_gn_relu<<<(n4 + 255) / 256, 256, 0, stream>>>((const v4f*)pre, stats,
                                                    gng, gnb, (v4f*)d_out);
}